// clique_69329362092377
// MI455X (gfx1250) — compile-verified
//
#include <hip/hip_runtime.h>
#include <math.h>

typedef __attribute__((ext_vector_type(2))) float v2f;
typedef __attribute__((ext_vector_type(8))) float v8f;

#define NN 100000
#define EE 3200000
#define NPICKN 50000

__device__ __forceinline__ v8f wmma4(v2f a, v2f b, v8f c) {
  // V_WMMA_F32_16X16X4_F32 : D(16x16,f32) = A(16x4,f32) x B(4x16,f32) + C
  return __builtin_amdgcn_wmma_f32_16x16x4_f32(false, a, false, b, (short)0, c,
                                               false, false);
}

// ---------------------------------------------------------------------------
// K1: y[N,128] = x[N,128] @ l0_W^T[128,128] + l0_b   (one wave per 16x16 tile)
// ---------------------------------------------------------------------------
__global__ __launch_bounds__(256) void gemm_x0(const float* __restrict__ x,
                                               const float* __restrict__ W,
                                               const float* __restrict__ bias,
                                               float* __restrict__ y) {
  int w = threadIdx.x >> 5, lane = threadIdx.x & 31;
  int node0 = blockIdx.x * 16;     // 6250 blocks exactly
  int col0 = w * 16;               // 8 waves -> 8 col tiles of 16
  int m = lane & 15, hi = lane >> 4, kb = hi * 2;
  const float* ar = x + (size_t)(node0 + m) * 128 + kb;
  const float* br = W + (size_t)(col0 + m) * 128 + kb;  // B[k][o] = W[o][k]
  v8f c = {};
#pragma unroll
  for (int k0 = 0; k0 < 128; k0 += 4) {
    v2f a = *(const v2f*)(ar + k0);
    v2f b = *(const v2f*)(br + k0);
    c = wmma4(a, b, c);
  }
  float bv = bias[col0 + m];
  int rb = node0 + hi * 8;
#pragma unroll
  for (int r = 0; r < 8; ++r)
    y[(size_t)(rb + r) * 128 + col0 + m] = c[r] + bv;
}

// ---------------------------------------------------------------------------
// K2: per-column sum / sumsq over y[N,128]
// ---------------------------------------------------------------------------
__global__ __launch_bounds__(256) void stats128(const float* __restrict__ y,
                                                float* __restrict__ sum,
                                                float* __restrict__ sumsq) {
  __shared__ float ls[256];
  int c = threadIdx.x & 127, half = threadIdx.x >> 7;
  int r0 = blockIdx.x * 256 + half;
  float s = 0.f, q = 0.f;
  for (int i = 0; i < 128; ++i) {
    int r = r0 + 2 * i;
    if (r < NN) {
      float v = y[(size_t)r * 128 + c];
      s += v;
      q += v * v;
    }
  }
  ls[threadIdx.x] = s;
  __syncthreads();
  if (half == 0) atomicAdd(&sum[c], ls[c] + ls[c + 128]);
  __syncthreads();
  ls[threadIdx.x] = q;
  __syncthreads();
  if (half == 0) atomicAdd(&sumsq[c], ls[c] + ls[c + 128]);
}

// ---------------------------------------------------------------------------
// K3: GraphNorm -> affine coeffs A,B per column
// out = w*(y - ms*mean)/sqrt(var+eps) + b = A*y + B
// var = E[y^2] - 2*ms*mean^2 + ms^2*mean^2
// ---------------------------------------------------------------------------
__global__ void norm_coeffs(const float* __restrict__ sum,
                            const float* __restrict__ sumsq,
                            const float* __restrict__ w,
                            const float* __restrict__ b,
                            const float* __restrict__ ms, float invn, int ncols,
                            float* __restrict__ A, float* __restrict__ B) {
  int c = blockIdx.x * blockDim.x + threadIdx.x;
  if (c >= ncols) return;
  float mean = sum[c] * invn;
  float ex2 = sumsq[c] * invn;
  float m = ms[c];
  float var = ex2 - 2.f * m * mean * mean + m * m * mean * mean;
  float inv = rsqrtf(var + 1e-5f);
  float a = w[c] * inv;
  A[c] = a;
  B[c] = b[c] - a * m * mean;
}

__global__ void apply_norm128(float* __restrict__ y,
                              const float* __restrict__ A,
                              const float* __restrict__ B) {
  int gid = blockIdx.x * 256 + threadIdx.x;
  if (gid >= NN * 128) return;
  int c = gid & 127;
  y[gid] = A[c] * y[gid] + B[c];
}

// ---------------------------------------------------------------------------
// K4: cur = x1 = relu(x0 @ l1_W^T + l1_b)  [N,128]->[N,10] via WMMA
// ---------------------------------------------------------------------------
__global__ __launch_bounds__(256) void x1_kernel(const float* __restrict__ x0,
                                                 const float* __restrict__ W,
                                                 const float* __restrict__ bias,
                                                 float* __restrict__ cur) {
  int wv = blockIdx.x * 8 + (threadIdx.x >> 5);
  if (wv >= NN / 16) return;  // wave-uniform guard
  int lane = threadIdx.x & 31;
  int m = lane & 15, hi = lane >> 4, kb = hi * 2;
  int node0 = wv * 16;
  const float* ar = x0 + (size_t)(node0 + m) * 128 + kb;
  bool valid = m < 10;
  const float* br = W + (size_t)(valid ? m : 0) * 128 + kb;
  v8f c = {};
#pragma unroll
  for (int k0 = 0; k0 < 128; k0 += 4) {
    v2f a = *(const v2f*)(ar + k0);
    v2f b;
    if (valid)
      b = *(const v2f*)(br + k0);
    else {
      b.x = 0.f;
      b.y = 0.f;
    }
    c = wmma4(a, b, c);
  }
  if (valid) {
    float bv = bias[m];
    int rb = node0 + hi * 8;
#pragma unroll
    for (int r = 0; r < 8; ++r)
      cur[(size_t)(rb + r) * 10 + m] = fmaxf(c[r] + bv, 0.f);
  }
}

// ---------------------------------------------------------------------------
// K5: column stats of edge pre-norm y_e = edge_attr @ l0e_W^T + b (not stored)
// ---------------------------------------------------------------------------
__global__ __launch_bounds__(256) void edge_stats(
    const float* __restrict__ ea, const float* __restrict__ W,
    const float* __restrict__ bias, float* __restrict__ sum,
    float* __restrict__ sumsq) {
  __shared__ float ls[16], lq[16];
  if (threadIdx.x < 16) {
    ls[threadIdx.x] = 0.f;
    lq[threadIdx.x] = 0.f;
  }
  __syncthreads();
  int wv = blockIdx.x * 8 + (threadIdx.x >> 5);  // 200000 tiles exactly
  int lane = threadIdx.x & 31;
  int m = lane & 15, hi = lane >> 4, kb = hi * 2;
  int e0 = wv * 16;
  const float* ar = ea + (size_t)(e0 + m) * 16 + kb;
  const float* br = W + (size_t)m * 16 + kb;
  v8f c = {};
#pragma unroll
  for (int k0 = 0; k0 < 16; k0 += 4) {
    v2f a = *(const v2f*)(ar + k0);
    v2f b = *(const v2f*)(br + k0);
    c = wmma4(a, b, c);
  }
  float bv = bias[m];
  float s = 0.f, q = 0.f;
#pragma unroll
  for (int r = 0; r < 8; ++r) {
    float v = c[r] + bv;
    s += v;
    q += v * v;
  }
  atomicAdd(&ls[m], s);
  atomicAdd(&lq[m], q);
  __syncthreads();
  if (threadIdx.x < 16) {
    atomicAdd(&sum[threadIdx.x], ls[threadIdx.x]);
    atomicAdd(&sumsq[threadIdx.x], lq[threadIdx.x]);
  }
}

// ---------------------------------------------------------------------------
// K6: fused edge MLP per 16-edge tile:
//   ea_n = An*(edge_attr@l0e_W^T + l0e_b) + Bn        (WMMA, staged in LDS)
//   eB   = ea_n @ eA_W^T + eA_b                       (WMMA)
//   t    = relu(ea_n @ e1_W^T + e1_b)                 (WMMA, restaged in LDS)
//   eA   = t @ e2_W^T + e2_b                          (WMMA)
// ---------------------------------------------------------------------------
__global__ __launch_bounds__(256) void edge_mlp(
    const float* __restrict__ ea, const float* __restrict__ W0,
    const float* __restrict__ b0, const float* __restrict__ An,
    const float* __restrict__ Bn, const float* __restrict__ WeA,
    const float* __restrict__ beA, const float* __restrict__ W1,
    const float* __restrict__ b1, const float* __restrict__ W2,
    const float* __restrict__ b2, float* __restrict__ outA,
    float* __restrict__ outB) {
  __shared__ float stage[8][16][17];
  int w = threadIdx.x >> 5, lane = threadIdx.x & 31;
  int wv = blockIdx.x * 8 + w;
  int m = lane & 15, hi = lane >> 4, kb = hi * 2;
  int e0 = wv * 16;
  float(*st)[17] = stage[w];

  // stage 1: normalized input projection
  {
    const float* ar = ea + (size_t)(e0 + m) * 16 + kb;
    const float* br = W0 + (size_t)m * 16 + kb;
    v8f c = {};
#pragma unroll
    for (int k0 = 0; k0 < 16; k0 += 4) {
      v2f a = *(const v2f*)(ar + k0);
      v2f b = *(const v2f*)(br + k0);
      c = wmma4(a, b, c);
    }
    float bv = b0[m], Ac = An[m], Bc = Bn[m];
#pragma unroll
    for (int r = 0; r < 8; ++r) st[r + 8 * hi][m] = Ac * (c[r] + bv) + Bc;
  }
  // stage 2: eB = ea_n @ eA_W^T + eA_b
  {
    v8f c = {};
#pragma unroll
    for (int k0 = 0; k0 < 16; k0 += 4) {
      v2f a, b;
      a.x = st[m][k0 + kb];
      a.y = st[m][k0 + kb + 1];
      b.x = (m < 10) ? WeA[m * 16 + k0 + kb] : 0.f;
      b.y = (m < 10) ? WeA[m * 16 + k0 + kb + 1] : 0.f;
      c = wmma4(a, b, c);
    }
    if (m < 10) {
      float bb = beA[m];
#pragma unroll
      for (int r = 0; r < 8; ++r)
        outB[(size_t)(e0 + r + 8 * hi) * 10 + m] = c[r] + bb;
    }
  }
  // stage 3: t = relu(ea_n @ e1_W^T + e1_b), restage
  {
    v8f c = {};
#pragma unroll
    for (int k0 = 0; k0 < 16; k0 += 4) {
      v2f a, b;
      a.x = st[m][k0 + kb];
      a.y = st[m][k0 + kb + 1];
      b.x = (m < 10) ? W1[m * 16 + k0 + kb] : 0.f;
      b.y = (m < 10) ? W1[m * 16 + k0 + kb + 1] : 0.f;
      c = wmma4(a, b, c);
    }
    float bt = (m < 10) ? b1[m] : 0.f;
#pragma unroll
    for (int r = 0; r < 8; ++r) {
      float v = (m < 10) ? fmaxf(c[r] + bt, 0.f) : 0.f;
      st[r + 8 * hi][m] = v;
    }
  }
  // stage 4: eA = t @ e2_W^T + e2_b   (K padded 10->16 with zeros)
  {
    v8f c = {};
#pragma unroll
    for (int k0 = 0; k0 < 16; k0 += 4) {
      v2f a, b;
      a.x = st[m][k0 + kb];
      a.y = st[m][k0 + kb + 1];
      b.x = (m < 10 && (k0 + kb) < 10) ? W2[m * 10 + k0 + kb] : 0.f;
      b.y = (m < 10 && (k0 + kb + 1) < 10) ? W2[m * 10 + k0 + kb + 1] : 0.f;
      c = wmma4(a, b, c);
    }
    if (m < 10) {
      float bb = b2[m];
#pragma unroll
      for (int r = 0; r < 8; ++r)
        outA[(size_t)(e0 + r + 8 * hi) * 10 + m] = c[r] + bb;
    }
  }
}

// ---------------------------------------------------------------------------
// K7: qkvs[N,40] = concat(x0,cur)[N,138] @ [Wq;Wk;Wv;Ws]^T + biases (WMMA)
// ---------------------------------------------------------------------------
__global__ __launch_bounds__(256) void qkvs_kernel(
    const float* __restrict__ x0, const float* __restrict__ cur,
    const float* __restrict__ Wq, const float* __restrict__ bq,
    const float* __restrict__ Wk, const float* __restrict__ bk,
    const float* __restrict__ Wv, const float* __restrict__ bv,
    const float* __restrict__ Ws, const float* __restrict__ bs,
    float* __restrict__ qkvs) {
  int wv = blockIdx.x * 8 + (threadIdx.x >> 5);
  if (wv >= (NN / 16) * 3) return;  // wave-uniform
  int nt = wv / 3, ct = wv - nt * 3;
  int lane = threadIdx.x & 31;
  int m = lane & 15, hi = lane >> 4, kb = hi * 2;
  int node0 = nt * 16;
  int oc = ct * 16 + m;  // 0..47, valid < 40
  const float* Wrow = Wq;
  float bias = 0.f;
  if (oc < 10) {
    Wrow = Wq + oc * 138;
    bias = bq[oc];
  } else if (oc < 20) {
    Wrow = Wk + (oc - 10) * 138;
    bias = bk[oc - 10];
  } else if (oc < 30) {
    Wrow = Wv + (oc - 20) * 138;
    bias = bv[oc - 20];
  } else if (oc < 40) {
    Wrow = Ws + (oc - 30) * 138;
    bias = bs[oc - 30];
  }
  const float* xr = x0 + (size_t)(node0 + m) * 128;
  const float* cr = cur + (size_t)(node0 + m) * 10;
  v8f c = {};
#pragma unroll
  for (int k0 = 0; k0 < 144; k0 += 4) {  // K=138 padded to 144
    int kk = k0 + kb;
    v2f a, b;
    a.x = (kk < 128) ? xr[kk] : ((kk < 138) ? cr[kk - 128] : 0.f);
    a.y = (kk + 1 < 128) ? xr[kk + 1] : ((kk + 1 < 138) ? cr[kk - 127] : 0.f);
    b.x = (oc < 40 && kk < 138) ? Wrow[kk] : 0.f;
    b.y = (oc < 40 && kk + 1 < 138) ? Wrow[kk + 1] : 0.f;
    c = wmma4(a, b, c);
  }
  if (oc < 40) {
    int rb = node0 + hi * 8;
#pragma unroll
    for (int r = 0; r < 8; ++r)
      qkvs[(size_t)(rb + r) * 40 + oc] = c[r] + bias;
  }
}

// ---------------------------------------------------------------------------
// Edge attention passes (thread per edge; ek recomputed, We cached in LDS)
// ---------------------------------------------------------------------------
__device__ __forceinline__ unsigned f2sort(float f) {
  unsigned u = __float_as_uint(f);
  return (u & 0x80000000u) ? ~u : (u | 0x80000000u);
}

__global__ __launch_bounds__(256) void edge_alpha(
    const int* __restrict__ src, const int* __restrict__ dst,
    const float* __restrict__ ef, const float* __restrict__ We,
    const float* __restrict__ qkvs, float* __restrict__ alpha,
    unsigned* __restrict__ amaxu) {
  __shared__ float w[100];
  if (threadIdx.x < 100) w[threadIdx.x] = We[threadIdx.x];
  __syncthreads();
  int e = blockIdx.x * 256 + threadIdx.x;
  if (e >= EE) return;
  int s = src[e], d = dst[e];
  float ev[10];
#pragma unroll
  for (int j = 0; j < 10; ++j) ev[j] = ef[(size_t)e * 10 + j];
  const float* q = qkvs + (size_t)d * 40;
  const float* k = qkvs + (size_t)s * 40 + 10;
  float a = 0.f;
#pragma unroll
  for (int h = 0; h < 10; ++h) {
    float ek = 0.f;
#pragma unroll
    for (int j = 0; j < 10; ++j) ek += ev[j] * w[h * 10 + j];
    a += q[h] * (k[h] + ek);
  }
  a *= 0.31622776601683794f;  // 1/sqrt(10)
  alpha[e] = a;
  atomicMax(&amaxu[d], f2sort(a));
}

__global__ void amax_fin(unsigned* __restrict__ buf) {
  int n = blockIdx.x * 256 + threadIdx.x;
  if (n >= NN) return;
  unsigned u = buf[n];
  u = (u & 0x80000000u) ? (u & 0x7FFFFFFFu) : ~u;
  float f = __uint_as_float(u);
  if (!(fabsf(f) < __builtin_inff())) f = 0.f;  // non-finite -> 0
  buf[n] = __float_as_uint(f);
}

__global__ __launch_bounds__(256) void edge_exp(const int* __restrict__ dst,
                                                const float* __restrict__ amaxf,
                                                float* __restrict__ alpha,
                                                float* __restrict__ denom) {
  int e = blockIdx.x * 256 + threadIdx.x;
  if (e >= EE) return;
  int d = dst[e];
  float ex = __expf(alpha[e] - amaxf[d]);
  alpha[e] = ex;
  atomicAdd(&denom[d], ex);
}

__global__ __launch_bounds__(256) void edge_msg(
    const int* __restrict__ src, const int* __restrict__ dst,
    const float* __restrict__ ef, const float* __restrict__ We,
    const float* __restrict__ qkvs, const float* __restrict__ alpha,
    const float* __restrict__ denom, float* __restrict__ accum) {
  __shared__ float w[100];
  if (threadIdx.x < 100) w[threadIdx.x] = We[threadIdx.x];
  __syncthreads();
  int e = blockIdx.x * 256 + threadIdx.x;
  if (e >= EE) return;
  int s = src[e], d = dst[e];
  float wt = alpha[e] / (denom[d] + 1e-16f);
  float ev[10];
#pragma unroll
  for (int j = 0; j < 10; ++j) ev[j] = ef[(size_t)e * 10 + j];
  const float* v = qkvs + (size_t)s * 40 + 20;
#pragma unroll
  for (int h = 0; h < 10; ++h) {
    float ek = 0.f;
#pragma unroll
    for (int j = 0; j < 10; ++j) ek += ev[j] * w[h * 10 + j];
    atomicAdd(&accum[(size_t)d * 10 + h], (v[h] + ek) * wt);
  }
}

__global__ void node_update(const float* __restrict__ accum,
                            const float* __restrict__ qkvs,
                            float* __restrict__ cur) {
  int gid = blockIdx.x * 256 + threadIdx.x;
  if (gid >= NN * 10) return;
  int n = gid / 10, h = gid - n * 10;
  cur[gid] = fmaxf(accum[gid] + qkvs[(size_t)n * 40 + 30 + h], 0.f);
}

__global__ void final_pick(const int* __restrict__ pick,
                           const float* __restrict__ accum,
                           const float* __restrict__ qkvs,
                           const float* __restrict__ linW,
                           const float* __restrict__ linb,
                           float* __restrict__ out) {
  int i = blockIdx.x * 256 + threadIdx.x;
  if (i >= NPICKN) return;
  int n = pick[i];
  float l0 = linb[0], l1 = linb[1];
#pragma unroll
  for (int j = 0; j < 10; ++j) {
    float h = accum[(size_t)n * 10 + j] + qkvs[(size_t)n * 40 + 30 + j];
    l0 += h * linW[j];
    l1 += h * linW[10 + j];
  }
  float mx = fmaxf(l0, l1);
  float e0 = __expf(l0 - mx), e1 = __expf(l1 - mx);
  float inv = 1.f / (e0 + e1);
  out[(size_t)i * 2] = e0 * inv;
  out[(size_t)i * 2 + 1] = e1 * inv;
}

// ---------------------------------------------------------------------------
extern "C" void kernel_launch(void* const* d_in, const int* in_sizes, int n_in,
                              void* d_out, int out_size, void* d_ws,
                              size_t ws_size, hipStream_t stream) {
  const float* x = (const float*)d_in[0];
  const int* eidx = (const int*)d_in[2];
  const int* src = eidx;
  const int* dst = eidx + EE;
  const float* edge_attr = (const float*)d_in[8];
  const int* pickable = (const int*)d_in[9];
  // params in setup_inputs() dict insertion order, starting at d_in[10]
  const float* P[56];
  for (int i = 0; i < 56; ++i) P[i] = (const float*)d_in[10 + i];
  const float *l0_W = P[0], *l0_b = P[1], *l0e_W = P[2], *l0e_b = P[3];
  const float *n_w = P[4], *n_b = P[5], *n_ms = P[6];
  const float *n2_w = P[7], *n2_b = P[8], *n2_ms = P[9];
  const float *e1_W = P[10], *e1_b = P[11], *e2_W = P[12], *e2_b = P[13];
  const float *eA_W = P[14], *eA_b = P[15], *l1_W = P[16], *l1_b = P[17];
  const float *lin_W = P[18], *lin_b = P[19];

  // workspace carve-out
  char* ws = (char*)d_ws;
  size_t off = 0;
  auto take = [&](size_t bytes) -> char* {
    char* p = ws + off;
    off += (bytes + 255) & ~(size_t)255;
    return p;
  };
  float* x0 = (float*)take((size_t)NN * 128 * 4);
  float* cur = (float*)take((size_t)NN * 10 * 4);
  float* qkvs = (float*)take((size_t)NN * 40 * 4);
  float* alpha = (float*)take((size_t)EE * 4);
  float* eAbuf = (float*)take((size_t)EE * 10 * 4);
  float* eBbuf = (float*)take((size_t)EE * 10 * 4);
  float* amax = (float*)take((size_t)NN * 4);   // contiguous with denom/accum
  float* denom = (float*)take((size_t)NN * 4);
  float* accum = (float*)take((size_t)NN * 10 * 4);
  float* nstat = (float*)take(128 * 4 * 4);  // sum, sumsq, A, B
  float* estat = (float*)take(16 * 4 * 4);
  float *nsum = nstat, *nsumsq = nstat + 128, *nA = nstat + 256,
        *nB = nstat + 384;
  float *esum = estat, *esumsq = estat + 16, *eAc = estat + 32,
        *eBc = estat + 48;
  size_t zbytes = (size_t)((char*)accum - (char*)amax) + (size_t)NN * 10 * 4;

  // ---- node path: x0 = GraphNorm(x @ l0_W^T + b); cur = x1 ----
  hipMemsetAsync(nsum, 0, 2 * 128 * 4, stream);
  hipMemsetAsync(esum, 0, 2 * 16 * 4, stream);
  gemm_x0<<<NN / 16, 256, 0, stream>>>(x, l0_W, l0_b, x0);
  stats128<<<(NN + 255) / 256, 256, 0, stream>>>(x0, nsum, nsumsq);
  norm_coeffs<<<1, 128, 0, stream>>>(nsum, nsumsq, n_w, n_b, n_ms, 1.f / NN,
                                     128, nA, nB);
  apply_norm128<<<(NN * 128 + 255) / 256, 256, 0, stream>>>(x0, nA, nB);
  x1_kernel<<<(NN / 16 + 7) / 8, 256, 0, stream>>>(x0, l1_W, l1_b, cur);

  // ---- edge path: eA, eB from GraphNorm(edge_attr @ l0e_W^T + b) ----
  edge_stats<<<EE / 16 / 8, 256, 0, stream>>>(edge_attr, l0e_W, l0e_b, esum,
                                              esumsq);
  norm_coeffs<<<1, 32, 0, stream>>>(esum, esumsq, n2_w, n2_b, n2_ms, 1.f / EE,
                                    16, eAc, eBc);
  edge_mlp<<<EE / 16 / 8, 256, 0, stream>>>(edge_attr, l0e_W, l0e_b, eAc, eBc,
                                            eA_W, eA_b, e1_W, e1_b, e2_W, e2_b,
                                            eAbuf, eBbuf);

  // ---- 4 TransformerConv layers: l2(eA), l3(eA), l4(eB), l5(eB) ----
  for (int layer = 0; layer < 4; ++layer) {
    const float* ef = (layer < 2) ? eAbuf : eBbuf;
    int lb = 20 + layer * 9;
    const float *Wq = P[lb], *bq = P[lb + 1], *Wk = P[lb + 2], *bk = P[lb + 3];
    const float *Wv = P[lb + 4], *bv = P[lb + 5], *We = P[lb + 6];
    const float *Ws = P[lb + 7], *bs = P[lb + 8];

    qkvs_kernel<<<((NN / 16) * 3 + 7) / 8, 256, 0, stream>>>(
        x0, cur, Wq, bq, Wk, bk, Wv, bv, Ws, bs, qkvs);
    hipMemsetAsync(amax, 0, zbytes, stream);  // amax, denom, accum
    edge_alpha<<<EE / 256, 256, 0, stream>>>(src, dst, ef, We, qkvs, alpha,
                                             (unsigned*)amax);
    amax_fin<<<(NN + 255) / 256, 256, 0, stream>>>((unsigned*)amax);
    edge_exp<<<EE / 256, 256, 0, stream>>>(dst, amax, alpha, denom);
    edge_msg<<<EE / 256, 256, 0, stream>>>(src, dst, ef, We, qkvs, alpha, denom,
                                           accum);
    if (layer < 3)
      node_update<<<(NN * 10 + 255) / 256, 256, 0, stream>>>(accum, qkvs, cur);
  }

  // ---- logits @ pickable + softmax ----
  final_pick<<<(NPICKN + 255) / 256, 256, 0, stream>>>(
      pickable, accum, qkvs, lin_W, lin_b, (float*)d_out);
}